// CAREGNN_26087631356718
// MI455X (gfx1250) — compile-verified
//
#include <hip/hip_runtime.h>
#include <hip/hip_bf16.h>
#include <math.h>

#define NF    128   // feature dim
#define EMBD  64    // emb dim
#define MAXD  16    // padded max degree
#define NCLS  2

typedef float v2f __attribute__((ext_vector_type(2)));
typedef float v8f __attribute__((ext_vector_type(8)));

// ---------------------------------------------------------------------------
// Phase 1: score_logits = x @ w_dist + b_dist  (per node), s0 = tanh(logit0)
// One wave32 per node; lanes split the 128 features, shfl-xor reduction.
// ---------------------------------------------------------------------------
__global__ __launch_bounds__(256)
void care_phase1(const float* __restrict__ x,
                 const float* __restrict__ w_dist,
                 const float* __restrict__ b_dist,
                 float* __restrict__ score_out,   // [N,2]
                 float* __restrict__ s0,          // [N]
                 int N)
{
    int wave = (int)((blockIdx.x * blockDim.x + threadIdx.x) >> 5);
    int lane = threadIdx.x & 31;
    if (wave >= N) return;
    const float* xr = x + (size_t)wave * NF;
    float acc0 = 0.f, acc1 = 0.f;
    #pragma unroll
    for (int f = 0; f < NF; f += 32) {
        float v = xr[f + lane];
        acc0 += v * w_dist[(f + lane) * 2 + 0];
        acc1 += v * w_dist[(f + lane) * 2 + 1];
    }
    #pragma unroll
    for (int off = 16; off > 0; off >>= 1) {
        acc0 += __shfl_xor(acc0, off, 32);
        acc1 += __shfl_xor(acc1, off, 32);
    }
    if (lane == 0) {
        float l0 = acc0 + b_dist[0];
        float l1 = acc1 + b_dist[1];
        score_out[(size_t)wave * 2 + 0] = l0;
        score_out[(size_t)wave * 2 + 1] = l1;
        s0[wave] = tanhf(l0);
    }
}

// ---------------------------------------------------------------------------
// Phase 2: neighbor filter + mean aggregate + fused SAGE GEMM (fp32 WMMA)
// Block = 128 threads = 4 waves; each wave owns one 16-node tile.
// C accumulates BOTH matmuls:  C += A_self*W_self + A_neigh*W_neigh.
// Both weight matrices are prepacked once per block into dynamic LDS in
// WMMA-fragment order: shW[(i*4+ct)*32+lane] = {ws_kk, ws_kk1, wn_kk, wn_kk1}
// so the hot loop needs only one ds_load_b128 per (K-step, col-tile).
// ---------------------------------------------------------------------------
__global__ __launch_bounds__(128)
void care_phase2(const float* __restrict__ x,
                 const float* __restrict__ w_self,
                 const float* __restrict__ w_neigh,
                 const float* __restrict__ b_sage,
                 const float* __restrict__ w_lin,
                 const float* __restrict__ b_lin,
                 const int*   __restrict__ neighbors,  // [N,MAXD]
                 const int*   __restrict__ degrees,    // [N]
                 const float* __restrict__ s0,         // [N]
                 float* __restrict__ logits_out,       // [N,2]
                 int N)
{
    extern __shared__ float4 shW[];            // [NF/4][4][32] fragments (64 KB)
    __shared__ float sh_an[4][16][NF + 4];     // per-wave mean-neighbor tile
    __shared__ float sh_h [4][16][EMBD + 2];   // per-wave hidden tile
    __shared__ float sh_wlin[EMBD * NCLS];
    __shared__ int   sh_sel[4][16][MAXD / 2];  // selected neighbor ids (k<=8)
    __shared__ int   sh_k [4][16];

    const int tid  = threadIdx.x;
    const int wid  = tid >> 5;
    const int lane = tid & 31;
    const int hi   = lane >> 4;      // 0/1 -> K sub-pair of the fragment
    const int lo   = lane & 15;      // row (A) / col (B,C)

    int tile0 = (int)blockIdx.x * 64 + wid * 16;
    const bool active = (tile0 < N);
    if (!active) tile0 = 0;          // keep wave resident for barriers

    // ---- prepack both weight matrices into LDS in fragment order -----------
    #pragma unroll 4
    for (int idx = tid; idx < (NF / 4) * 4 * 32; idx += 128) {
        const int l   = idx & 31;
        const int ct  = (idx >> 5) & 3;
        const int i   = idx >> 7;
        const int kk  = 4 * i + 2 * (l >> 4);
        const int col = ct * 16 + (l & 15);
        float4 w;
        w.x = w_self [(size_t)(kk)     * EMBD + col];
        w.y = w_self [(size_t)(kk + 1) * EMBD + col];
        w.z = w_neigh[(size_t)(kk)     * EMBD + col];
        w.w = w_neigh[(size_t)(kk + 1) * EMBD + col];
        shW[idx] = w;
    }
    if (tid < EMBD * NCLS) sh_wlin[tid] = w_lin[tid];

    // ---- stable k-smallest neighbor selection: lanes 0..15, one node each ---
    if (lane < 16) {
        int node = tile0 + lane;
        if (node >= N) node = N - 1;           // dummy (store masked later)
        int deg = degrees[node];
        if (deg < 1) deg = 1;
        if (deg > MAXD) deg = MAXD;
        int k = deg >> 1;
        if (k < 1) k = 1;
        float sc = s0[node];
        float dist[MAXD];
        int   ids [MAXD];
        #pragma unroll
        for (int j = 0; j < MAXD; ++j) {
            int nb = neighbors[(size_t)node * MAXD + j];
            ids[j] = nb;
            float d = fabsf(sc - s0[nb]);
            dist[j] = (j < deg) ? d : INFINITY; // padding sorts last
        }
        for (int t = 0; t < k; ++t) {          // stable: strict < keeps lowest idx
            int best = 0; float bv = INFINITY;
            #pragma unroll
            for (int j = 0; j < MAXD; ++j) {
                if (dist[j] < bv) { bv = dist[j]; best = j; }
            }
            sh_sel[wid][lane][t] = ids[best];
            dist[best] = INFINITY;
        }
        sh_k[wid][lane] = k;
    }
    __syncthreads();

    // ---- masked mean aggregation of selected neighbor features -> LDS ------
    // lane covers features {lane, lane+32, lane+64, lane+96}; loop count is
    // wave-uniform (all lanes read the same sh_k[wid][m]).
    for (int m = 0; m < 16; ++m) {
        int k = sh_k[wid][m];
        float a0 = 0.f, a1 = 0.f, a2 = 0.f, a3 = 0.f;
        for (int j = 0; j < k; ++j) {
            const float* xr = x + (size_t)sh_sel[wid][m][j] * NF;
            a0 += xr[lane];
            a1 += xr[lane + 32];
            a2 += xr[lane + 64];
            a3 += xr[lane + 96];
        }
        float inv = 1.0f / (float)k;
        sh_an[wid][m][lane]      = a0 * inv;
        sh_an[wid][m][lane + 32] = a1 * inv;
        sh_an[wid][m][lane + 64] = a2 * inv;
        sh_an[wid][m][lane + 96] = a3 * inv;
    }
    __syncthreads();

    // ---- fused fp32 WMMA GEMM: h = relu(x@Ws + mean@Wn + b) ----------------
    // A 16x4 f32 layout: lanes 0-15 hold row M=lane,    K=k0+{0,1};
    //                    lanes 16-31 hold row M=lane-16, K=k0+{2,3}.
    // B 4x16 f32 layout (symmetric): prepacked in shW.
    int srow = tile0 + lo;
    if (srow >= N) srow = N - 1;
    const float* xself = x + (size_t)srow * NF;

    v8f c[4];
    #pragma unroll
    for (int ct = 0; ct < 4; ++ct) c[ct] = (v8f){0.f,0.f,0.f,0.f,0.f,0.f,0.f,0.f};

    for (int i = 0; i < NF / 4; ++i) {
        const int kk = 4 * i + hi * 2;
        v2f a_s, a_n;
        a_s.x = xself[kk];          a_s.y = xself[kk + 1];
        a_n.x = sh_an[wid][lo][kk]; a_n.y = sh_an[wid][lo][kk + 1];
        #pragma unroll
        for (int ct = 0; ct < 4; ++ct) {
            float4 wf = shW[(i * 4 + ct) * 32 + lane];   // ds_load_b128
            v2f b_s; b_s.x = wf.x; b_s.y = wf.y;
            v2f b_n; b_n.x = wf.z; b_n.y = wf.w;
            c[ct] = __builtin_amdgcn_wmma_f32_16x16x4_f32(
                        false, a_s, false, b_s, (short)0, c[ct], false, false);
            c[ct] = __builtin_amdgcn_wmma_f32_16x16x4_f32(
                        false, a_n, false, b_n, (short)0, c[ct], false, false);
        }
    }

    // C layout: c[ct][j] = value at row (j + 8*hi), col (ct*16 + lo)
    #pragma unroll
    for (int ct = 0; ct < 4; ++ct) {
        const int e = ct * 16 + lo;
        const float be = b_sage[e];
        #pragma unroll
        for (int j = 0; j < 8; ++j) {
            float v = c[ct][j] + be;
            sh_h[wid][j + 8 * hi][e] = v > 0.f ? v : 0.f;
        }
    }
    __syncthreads();

    // ---- final linear: logits = h @ w_lin + b_lin (one output per lane) ----
    {
        const int m = lane >> 1, cls = lane & 1;
        float acc = b_lin[cls];
        #pragma unroll 8
        for (int e = 0; e < EMBD; ++e)
            acc += sh_h[wid][m][e] * sh_wlin[e * NCLS + cls];
        const int node = tile0 + m;
        if (active && node < N)
            logits_out[(size_t)node * NCLS + cls] = acc;
    }
}

// ---------------------------------------------------------------------------
extern "C" void kernel_launch(void* const* d_in, const int* in_sizes, int n_in,
                              void* d_out, int out_size, void* d_ws, size_t ws_size,
                              hipStream_t stream) {
    const float* x       = (const float*)d_in[0];
    const float* w_dist  = (const float*)d_in[1];
    const float* b_dist  = (const float*)d_in[2];
    const float* w_self  = (const float*)d_in[3];
    const float* w_neigh = (const float*)d_in[4];
    const float* b_sage  = (const float*)d_in[5];
    const float* w_lin   = (const float*)d_in[6];
    const float* b_lin   = (const float*)d_in[7];
    const int*   neigh   = (const int*)d_in[8];
    const int*   degs    = (const int*)d_in[9];

    const int N = in_sizes[9];                 // degrees: [N]
    float* out       = (float*)d_out;
    float* logits    = out;                    // [N,2] first output
    float* score_out = out + (size_t)N * NCLS; // [N,2] second output
    float* s0        = (float*)d_ws;           // N floats scratch

    // Phase 1: one wave per node (8 waves per 256-thread block)
    {
        int waves_per_block = 256 / 32;
        int blocks = (N + waves_per_block - 1) / waves_per_block;
        care_phase1<<<blocks, 256, 0, stream>>>(x, w_dist, b_dist,
                                                score_out, s0, N);
    }
    // Phase 2: 4 tiles of 16 nodes per 128-thread block,
    // 64 KB dynamic LDS for the prepacked weight fragments.
    {
        int blocks = (N + 63) / 64;
        size_t dynbytes = (size_t)(NF / 4) * 4 * 32 * sizeof(float4); // 65536
        care_phase2<<<blocks, 128, dynbytes, stream>>>(x, w_self, w_neigh,
                                                       b_sage, w_lin, b_lin,
                                                       neigh, degs, s0,
                                                       logits, N);
    }
}